// DistillSTU_59940563583686
// MI455X (gfx1250) — compile-verified
//
#include <hip/hip_runtime.h>

#define SEQ   2048
#define DM    768
#define KF    24
#define ST    100
#define STP   128   // padded state count (4 per lane * 32 lanes)
#define PROW  36    // padded transpose row (36*4 = 144 B -> float4 aligned)

typedef float v2f __attribute__((ext_vector_type(2)));
typedef float v8f __attribute__((ext_vector_type(8)));

// Broadcast lane j's value through an SGPR (j must be a literal after unroll).
__device__ __forceinline__ float bcast(float v, int j) {
    return __uint_as_float(__builtin_amdgcn_readlane(__float_as_uint(v), j));
}

// ---------------------------------------------------------------------------
// Kernel 1: xp = x @ M_inputs via V_WMMA_F32_16X16X4_F32, stored TRANSPOSED
// into u[d][s] so the scan kernel reads channel-major, coalesced.
// Wave layout per ISA 7.12.2 (f32):
//   A 16x4 : grp=l>>4, idx=l&15 ; VGPR0 = A[idx][k0+2*grp], VGPR1 = +1
//   B 4x16 : VGPR0 = B[k0+2*grp][idx], VGPR1 = B[k0+2*grp+1][idx]
//   C/D    : VGPR r = C[r + 8*grp][idx]  (8 s-contiguous floats per lane)
// ---------------------------------------------------------------------------
__global__ __launch_bounds__(128)
void stu_gemm_wmma(const float* __restrict__ x,      // (SEQ, DM)
                   const float* __restrict__ Mi,     // (DM, DM)
                   float* __restrict__ u)            // (DM, SEQ)
{
    const int wave = threadIdx.x >> 5;
    const int lane = threadIdx.x & 31;
    const int grp  = lane >> 4;
    const int idx  = lane & 15;

    const int sBase = (blockIdx.x * 4 + wave) * 16;
    const int dBase = blockIdx.y * 16;

    const float* __restrict__ xrow = x + (size_t)(sBase + idx) * DM;

    v8f c = {};
#pragma unroll 4
    for (int k0 = 0; k0 < DM; k0 += 4) {
        const int kk = k0 + 2 * grp;
        v2f a, b;
        const float2 av = *reinterpret_cast<const float2*>(xrow + kk);
        a.x = av.x; a.y = av.y;
        b.x = Mi[(size_t)kk * DM + dBase + idx];
        b.y = Mi[(size_t)(kk + 1) * DM + dBase + idx];
        c = __builtin_amdgcn_wmma_f32_16x16x4_f32(
                false, a, false, b, (short)0, c, false, false);
    }

    float* dst = u + (size_t)(dBase + idx) * SEQ + sBase + 8 * grp;
    float4 lo, hi;
    lo.x = c[0]; lo.y = c[1]; lo.z = c[2]; lo.w = c[3];
    hi.x = c[4]; hi.y = c[5]; hi.z = c[6]; hi.w = c[7];
    *reinterpret_cast<float4*>(dst)     = lo;
    *reinterpret_cast<float4*>(dst + 4) = hi;
}

// ---------------------------------------------------------------------------
// Kernel 2: fold C, Dv, Bm, M_filters into per-channel scan weights.
//   w[d][i]  = Bm[i] * sum_k (C[i][k] + C[i][k+24]) * Mf[k][d]   (i < 100)
//   dv_eff[d]=          sum_k (Dv[k]  + Dv[k+24])  * Mf[k][d]
// ---------------------------------------------------------------------------
__global__ __launch_bounds__(256)
void stu_prep(const float* __restrict__ C,    // (ST, 2K)
              const float* __restrict__ Dv,   // (2K)
              const float* __restrict__ Mf,   // (KF, DM)
              const float* __restrict__ Bm,   // (ST)
              float* __restrict__ w,          // (DM, STP)
              float* __restrict__ dveff)      // (DM)
{
    const int gid = blockIdx.x * blockDim.x + threadIdx.x;
    const int d = gid % DM;
    const int i = gid / DM;
    if (i >= STP) return;

    float acc = 0.f;
    if (i < ST) {
#pragma unroll
        for (int k = 0; k < KF; ++k)
            acc = __builtin_fmaf(C[i * (2 * KF) + k] + C[i * (2 * KF) + KF + k],
                                 Mf[k * DM + d], acc);
        acc *= Bm[i];
    }
    w[(size_t)d * STP + i] = acc;

    if (i == 0) {
        float dacc = 0.f;
#pragma unroll
        for (int k = 0; k < KF; ++k)
            dacc = __builtin_fmaf(Dv[k] + Dv[KF + k], Mf[k * DM + d], dacc);
        dveff[d] = dacc;
    }
}

// ---------------------------------------------------------------------------
// Kernel 3: diagonal-LDS scan. One wave32 per channel (4 channels / block).
// Lane owns states {lane, lane+32, lane+64, lane+96} (zero-padded past 100).
// Per step: v_readlane broadcast of u[t] (SGPR, no DS traffic), 4 FMA state
// update, 4 FMA partial dot, one ds_store of the partial. After 32 steps a
// per-wave LDS transpose turns 32 butterfly reductions into 8 float4 loads
// + local tree sum per lane. Lane l reduces step l, whose u value is lane
// l's own coalesced load -> Dv feed-through folds in for free.
// ---------------------------------------------------------------------------
__global__ __launch_bounds__(128)
void stu_scan(const float* __restrict__ u,      // (DM, SEQ)
              const float* __restrict__ A,      // (ST)
              const float* __restrict__ w,      // (DM, STP)
              const float* __restrict__ dveff,  // (DM)
              float* __restrict__ out)          // (SEQ, DM)
{
    const int tid  = threadIdx.x;
    const int wv   = tid >> 5;
    const int lane = tid & 31;
    const int d0   = blockIdx.x * 4;
    const int d    = d0 + wv;

    float a0 = (lane +  0 < ST) ? A[lane +  0] : 0.f;
    float a1 = (lane + 32 < ST) ? A[lane + 32] : 0.f;
    float a2 = (lane + 64 < ST) ? A[lane + 64] : 0.f;
    float a3 = (lane + 96 < ST) ? A[lane + 96] : 0.f;

    const float* wd = w + (size_t)d * STP;
    const float w0 = wd[lane], w1 = wd[lane + 32], w2 = wd[lane + 64], w3 = wd[lane + 96];
    const float dv = dveff[d];

    float g0 = 0.f, g1 = 0.f, g2 = 0.f, g3 = 0.f;

    __shared__ float pbuf[4][32][PROW];   // per-wave partial transpose buffer
    __shared__ float obuf[4][33];         // cross-wave output staging

    const float* __restrict__ uch = u + (size_t)d * SEQ;
    float v = uch[lane];                  // first chunk (coalesced)

    for (int t0 = 0; t0 < SEQ; t0 += 32) {
        // prefetch next chunk while this one computes
        float vn = 0.f;
        if (t0 + 32 < SEQ) vn = uch[t0 + 32 + lane];

#pragma unroll
        for (int j = 0; j < 32; ++j) {
            const float uj = bcast(v, j);          // SGPR broadcast of u[t0+j]
            g0 = __builtin_fmaf(a0, g0, uj);
            g1 = __builtin_fmaf(a1, g1, uj);
            g2 = __builtin_fmaf(a2, g2, uj);
            g3 = __builtin_fmaf(a3, g3, uj);
            float p = g0 * w0;
            p = __builtin_fmaf(g1, w1, p);
            p = __builtin_fmaf(g2, w2, p);
            p = __builtin_fmaf(g3, w3, p);
            pbuf[wv][j][lane] = p;                 // conflict-free b32 store
        }

        // transpose-reduce: lane l sums its contiguous row (step t0+l)
        const float4* prow = reinterpret_cast<const float4*>(&pbuf[wv][lane][0]);
        float4 q0 = prow[0], q1 = prow[1], q2 = prow[2], q3 = prow[3];
        float4 q4 = prow[4], q5 = prow[5], q6 = prow[6], q7 = prow[7];
        float s01 = ((q0.x + q0.y) + (q0.z + q0.w)) + ((q1.x + q1.y) + (q1.z + q1.w));
        float s23 = ((q2.x + q2.y) + (q2.z + q2.w)) + ((q3.x + q3.y) + (q3.z + q3.w));
        float s45 = ((q4.x + q4.y) + (q4.z + q4.w)) + ((q5.x + q5.y) + (q5.z + q5.w));
        float s67 = ((q6.x + q6.y) + (q6.z + q6.w)) + ((q7.x + q7.y) + (q7.z + q7.w));
        const float sum = (s01 + s23) + (s45 + s67);
        const float o = __builtin_fmaf(dv, v, sum);   // lane l's v == u[t0+l]

        // coalesced float4 rows out[t0..t0+31][d0..d0+3]
        obuf[wv][lane] = o;
        __syncthreads();
        if (tid < 32) {
            float4 ov;
            ov.x = obuf[0][tid]; ov.y = obuf[1][tid];
            ov.z = obuf[2][tid]; ov.w = obuf[3][tid];
            *reinterpret_cast<float4*>(&out[(size_t)(t0 + tid) * DM + d0]) = ov;
        }
        __syncthreads();

        v = vn;
    }
}

// ---------------------------------------------------------------------------
extern "C" void kernel_launch(void* const* d_in, const int* in_sizes, int n_in,
                              void* d_out, int out_size, void* d_ws, size_t ws_size,
                              hipStream_t stream) {
    (void)in_sizes; (void)n_in; (void)out_size; (void)ws_size;
    const float* x   = (const float*)d_in[0];
    /* d_in[1] = input_pos (unused) */
    const float* Mi  = (const float*)d_in[2];
    const float* Mf  = (const float*)d_in[3];
    const float* A   = (const float*)d_in[4];
    const float* Bm  = (const float*)d_in[5];
    const float* C   = (const float*)d_in[6];
    const float* Dv  = (const float*)d_in[7];
    float* out = (float*)d_out;

    char* ws = (char*)d_ws;
    float* u     = (float*)ws;                                        // DM*SEQ
    float* w     = (float*)(ws + (size_t)DM * SEQ * sizeof(float));   // DM*STP
    float* dveff = (float*)(ws + ((size_t)DM * SEQ + (size_t)DM * STP) * sizeof(float));

    stu_prep<<<(DM * STP + 255) / 256, 256, 0, stream>>>(C, Dv, Mf, Bm, w, dveff);

    dim3 ggrid(SEQ / 64, DM / 16);
    stu_gemm_wmma<<<ggrid, 128, 0, stream>>>(x, Mi, u);

    stu_scan<<<DM / 4, 128, 0, stream>>>(u, A, w, dveff, out);
}